// GIN_86045374808622
// MI455X (gfx1250) — compile-verified
//
#include <hip/hip_runtime.h>

typedef float v2f __attribute__((ext_vector_type(2)));
typedef float v8f __attribute__((ext_vector_type(8)));

#define GIN_N 50000
#define GIN_E 800000
#define GIN_F 128
#define GIN_C 32
#define BN_EPS 1e-5f
#define SCAN_BLOCK 1024

// ---------------------------------------------------------------- utilities
__global__ void zero_i_kernel(int* p, int n) {
  int i = blockIdx.x * blockDim.x + threadIdx.x;
  if (i < n) p[i] = 0;
}

__global__ void zero_f_kernel(float* p, int n) {
  int i = blockIdx.x * blockDim.x + threadIdx.x;
  if (i < n) p[i] = 0.0f;
}

// -------------------------------------------------------- CSR build (once)
// 1) histogram of in-degrees
__global__ void hist_kernel(const int* __restrict__ dst, int* __restrict__ deg,
                            int n) {
  int i = blockIdx.x * blockDim.x + threadIdx.x;
  const int stride = gridDim.x * blockDim.x;
  for (; i < n; i += stride) atomicAdd(&deg[dst[i]], 1);
}

// 2) exclusive scan over deg -> rowptr[0..n] (single block, LDS Hillis-Steele)
__global__ void exclusive_scan_kernel(const int* __restrict__ in,
                                      int* __restrict__ out, int n) {
  __shared__ int buf[SCAN_BLOCK];
  __shared__ int carry_s;
  if (threadIdx.x == 0) carry_s = 0;
  __syncthreads();
  for (int base = 0; base < n; base += SCAN_BLOCK) {
    const int i = base + threadIdx.x;
    const int v = (i < n) ? in[i] : 0;
    buf[threadIdx.x] = v;
    __syncthreads();
    for (int off = 1; off < SCAN_BLOCK; off <<= 1) {
      const int t = (threadIdx.x >= off) ? buf[threadIdx.x - off] : 0;
      __syncthreads();
      buf[threadIdx.x] += t;
      __syncthreads();
    }
    const int incl = buf[threadIdx.x];
    const int c = carry_s;
    if (i < n) out[i] = c + incl - v;  // exclusive
    __syncthreads();
    if (threadIdx.x == SCAN_BLOCK - 1) carry_s = c + incl;
    __syncthreads();
  }
  if (threadIdx.x == 0) out[n] = carry_s;
}

// 3) scatter source ids into CSR slots (integer atomics on cursors only)
__global__ void fill_kernel(const int* __restrict__ src,
                            const int* __restrict__ dst,
                            int* __restrict__ cursor, int* __restrict__ nbr,
                            int n) {
  int i = blockIdx.x * blockDim.x + threadIdx.x;
  const int stride = gridDim.x * blockDim.x;
  for (; i < n; i += stride) {
    const int p = atomicAdd(&cursor[dst[i]], 1);
    nbr[p] = src[i];
  }
}

// ---------------------------------- GIN aggregation: agg[v] = h[v] + sum h[u]
// Pure gather via CSR: no float atomics, b128 loads, one plain store.
// 128 threads = 4 nodes; 32 lanes x float4 cover the 128 features.
__global__ void gin_gather_kernel(const float* __restrict__ h,
                                  const int* __restrict__ rowptr,
                                  const int* __restrict__ nbr,
                                  float* __restrict__ agg, int nnodes) {
  const int node = blockIdx.x * 4 + (threadIdx.x >> 5);
  if (node >= nnodes) return;
  const int lane = threadIdx.x & 31;
  const float4* __restrict__ hv = (const float4*)h;

  float4 s = hv[(size_t)node * (GIN_F / 4) + lane];  // self term (eps=0)
  const int beg = rowptr[node];
  const int end = rowptr[node + 1];
  for (int j = beg; j < end; ++j) {
    const int u = nbr[j];
    const float4 t = hv[(size_t)u * (GIN_F / 4) + lane];
    s.x += t.x; s.y += t.y; s.z += t.z; s.w += t.w;
  }
  ((float4*)agg)[(size_t)node * (GIN_F / 4) + lane] = s;
}

// ------------------------------------------------ fp32 WMMA GEMM: Y = X*W + b
// X: nrows x kdim (row-major), W: kdim x ncols (row-major), Y: nrows x ncols.
// 256 threads = 8 wave32; each wave computes a 16x16 tile with
// V_WMMA_F32_16X16X4_F32. nrows%16==0, kdim%4==0, ncols%16==0 so EXEC is
// all-ones around every WMMA.
__global__ void gemm_bias_wmma_kernel(const float* __restrict__ X,
                                      const float* __restrict__ W,
                                      const float* __restrict__ bias,
                                      float* __restrict__ Y,
                                      int kdim, int ncols) {
  const int wave = threadIdx.x >> 5;
  const int lane = threadIdx.x & 31;
  const int col0 = wave * 16;
  if (col0 >= ncols) return;  // wave-uniform exit (ncols=32 case)

  const int row0 = blockIdx.x * 16;
  const int laneLo = lane & 15;   // M (for A) / N (for B,C,D)
  const int laneHi = lane >> 4;   // selects K pair / M+8 half

  // A frag: lane holds X[row0+laneLo][kb + 2*laneHi + {0,1}]  (8B-aligned)
  const float* __restrict__ xr =
      X + (size_t)(row0 + laneLo) * kdim + 2 * laneHi;
  // B frag: lane holds W[kb + 2*laneHi + {0,1}][col0 + laneLo]
  const float* __restrict__ wp =
      W + (size_t)(2 * laneHi) * ncols + col0 + laneLo;

  v8f acc = {};
  for (int kb = 0; kb < kdim; kb += 4) {
    v2f a = *(const v2f*)(xr + kb);
    v2f b;
    b.x = wp[(size_t)kb * ncols];
    b.y = wp[(size_t)(kb + 1) * ncols];
    acc = __builtin_amdgcn_wmma_f32_16x16x4_f32(
        /*neg_a=*/false, a, /*neg_b=*/false, b,
        /*c_mod=*/(short)0, acc, /*reuse_a=*/false, /*reuse_b=*/false);
  }

  const float bv = bias[col0 + laneLo];
#pragma unroll
  for (int j = 0; j < 8; ++j) {
    const int row = row0 + j + 8 * laneHi;  // C/D layout: VGPR j -> M=j (+8)
    Y[(size_t)row * ncols + col0 + laneLo] = acc[j] + bv;
  }
}

// -------------------------------------------------------- BN column reduction
__global__ void bn_stats_kernel(const float* __restrict__ Y,
                                float* __restrict__ stats,  // [2*ncols]
                                int nrows, int ncols) {
  const int col = threadIdx.x;
  float s = 0.0f, ss = 0.0f;
  for (int r = blockIdx.x; r < nrows; r += gridDim.x) {
    const float v = Y[(size_t)r * ncols + col];
    s += v;
    ss += v * v;
  }
  atomicAdd(&stats[col], s);
  atomicAdd(&stats[ncols + col], ss);
}

// ------------------------------------------------- BN normalize+affine(+ReLU)
__global__ void bn_apply_kernel(const float* Yin, const float* stats,
                                const float* g, const float* be, float* Yout,
                                int nrows, int ncols, int relu) {
  const int col = threadIdx.x;
  const float invN = 1.0f / (float)nrows;
  const float mean = stats[col] * invN;
  const float var = stats[ncols + col] * invN - mean * mean;  // biased var
  const float scale = g[col] * rsqrtf(var + BN_EPS);
  const float shift = be[col] - mean * scale;
  for (int r = blockIdx.x; r < nrows; r += gridDim.x) {
    float v = Yin[(size_t)r * ncols + col] * scale + shift;
    if (relu) v = fmaxf(v, 0.0f);
    Yout[(size_t)r * ncols + col] = v;
  }
}

// --------------------------------------------------------------------- driver
extern "C" void kernel_launch(void* const* d_in, const int* in_sizes, int n_in,
                              void* d_out, int out_size, void* d_ws,
                              size_t ws_size, hipStream_t stream) {
  (void)in_sizes; (void)n_in; (void)out_size; (void)ws_size;

  const float* x  = (const float*)d_in[0];
  const int* src  = (const int*)d_in[1];
  const int* dst  = (const int*)d_in[2];

  // Workspace layout (all 4-byte aligned):
  float* agg   = (float*)d_ws;                       // N*F
  float* ybuf  = agg + (size_t)GIN_N * GIN_F;        // N*F
  float* stats = ybuf + (size_t)GIN_N * GIN_F;       // 256
  int* deg     = (int*)(stats + 256);                // N
  int* rowptr  = deg + GIN_N;                        // N+1
  int* cursor  = rowptr + (GIN_N + 1);               // N
  int* nbr     = cursor + GIN_N;                     // E

  // ---- Build CSR adjacency (dst-grouped) once; reused by all 3 layers.
  zero_i_kernel<<<(GIN_N + 255) / 256, 256, 0, stream>>>(deg, GIN_N);
  hist_kernel<<<1024, 256, 0, stream>>>(dst, deg, GIN_E);
  exclusive_scan_kernel<<<1, SCAN_BLOCK, 0, stream>>>(deg, rowptr, GIN_N);
  hipMemcpyAsync(cursor, rowptr, GIN_N * sizeof(int),
                 hipMemcpyDeviceToDevice, stream);
  fill_kernel<<<1024, 256, 0, stream>>>(src, dst, cursor, nbr, GIN_E);

  const float* cur = x;  // current node features (never mutate d_in)
  for (int li = 0; li < 3; ++li) {
    const float* w1  = (const float*)d_in[3 + li * 8 + 0];
    const float* b1  = (const float*)d_in[3 + li * 8 + 1];
    const float* g1  = (const float*)d_in[3 + li * 8 + 2];
    const float* be1 = (const float*)d_in[3 + li * 8 + 3];
    const float* w2  = (const float*)d_in[3 + li * 8 + 4];
    const float* b2  = (const float*)d_in[3 + li * 8 + 5];
    const float* g2  = (const float*)d_in[3 + li * 8 + 6];
    const float* be2 = (const float*)d_in[3 + li * 8 + 7];
    const int dout = (li == 2) ? GIN_C : GIN_F;
    const int last = (li == 2);

    // 1) agg[v] = cur[v] + sum_{u->v} cur[u]   (pure gather, no f32 atomics)
    gin_gather_kernel<<<(GIN_N + 3) / 4, 128, 0, stream>>>(cur, rowptr, nbr,
                                                           agg, GIN_N);

    // 2) y = agg @ w1 + b1 ; BN1 ; inner ReLU  (in place in ybuf)
    gemm_bias_wmma_kernel<<<GIN_N / 16, 256, 0, stream>>>(agg, w1, b1, ybuf,
                                                          GIN_F, GIN_F);
    zero_f_kernel<<<1, 256, 0, stream>>>(stats, 256);
    bn_stats_kernel<<<512, GIN_F, 0, stream>>>(ybuf, stats, GIN_N, GIN_F);
    bn_apply_kernel<<<1024, GIN_F, 0, stream>>>(ybuf, stats, g1, be1, ybuf,
                                                GIN_N, GIN_F, /*relu=*/1);

    // 3) z = y @ w2 + b2 -> agg (reuse) ; BN2 (+outer ReLU except last layer)
    gemm_bias_wmma_kernel<<<GIN_N / 16, 256, 0, stream>>>(ybuf, w2, b2, agg,
                                                          GIN_F, dout);
    zero_f_kernel<<<1, 256, 0, stream>>>(stats, 256);
    bn_stats_kernel<<<512, dout, 0, stream>>>(agg, stats, GIN_N, dout);
    float* nxt = last ? (float*)d_out : ybuf;
    bn_apply_kernel<<<1024, dout, 0, stream>>>(agg, stats, g2, be2, nxt, GIN_N,
                                               dout, /*relu=*/last ? 0 : 1);
    cur = ybuf;  // layer output becomes next input
  }
}